// ksparse_38388417692284
// MI455X (gfx1250) — compile-verified
//
#include <hip/hip_runtime.h>

// k-sparse row masking: theta = row's 1843rd-smallest value (ascending sort
// index int((1-0.1)*2048) = 1843), out = x * (x > theta).
// One 256-thread block (8 wave32 waves) per row; exact 4x8-bit radix select.
// Row staged via CDNA5 async global->LDS (ASYNCcnt) path; 256-bin scan done
// as wave32 shuffle scan + 8-entry cross-wave combine (minimal barriers).

#define BLOCK 256
#define DIM 2048
#define VEC_PER_ROW (DIM / 4)                // 512 float4 per row
#define VEC_PER_THREAD (VEC_PER_ROW / BLOCK) // 2 float4 per thread
#define K_RANK 1843u
#define NWAVES (BLOCK / 32)

typedef int v4i __attribute__((ext_vector_type(4)));

__device__ __forceinline__ unsigned order_key(unsigned b) {
    // Monotone bijection float-bits -> u32 (strictly order preserving).
    return b ^ (((unsigned)((int)b >> 31)) | 0x80000000u);
}

__global__ __launch_bounds__(BLOCK) void ksparse_rowselect(
    const float* __restrict__ x, float* __restrict__ out) {
    __shared__ float4   srow[VEC_PER_ROW];   // 8 KB row staging (async target)
    __shared__ unsigned hist[BLOCK];
    __shared__ unsigned wtot[NWAVES];
    __shared__ unsigned sel[2];              // [0]=digit, [1]=new rank

    const int row  = blockIdx.x;
    const int tid  = threadIdx.x;
    const int lane = tid & 31;
    const int wave = tid >> 5;
    const float4* grow = (const float4*)(x + (size_t)row * DIM);

#if __has_builtin(__builtin_amdgcn_global_load_async_to_lds_b128)
    // CDNA5 async copy: global -> LDS without VGPR staging (ASYNCcnt path).
#pragma unroll
    for (int i = 0; i < VEC_PER_THREAD; ++i) {
        const int v = tid + i * BLOCK;
        __builtin_amdgcn_global_load_async_to_lds_b128(
            (v4i*)(grow + v), (v4i*)(srow + v), 0, 0);
    }
#if __has_builtin(__builtin_amdgcn_s_wait_asynccnt)
    __builtin_amdgcn_s_wait_asynccnt(0);
#else
    asm volatile("s_wait_asynccnt 0" ::: "memory");
#endif
    __syncthreads();                         // all waves' copies complete
    float4 a = srow[tid];
    float4 b = srow[tid + BLOCK];
#else
    float4 a = grow[tid];
    float4 b = grow[tid + BLOCK];
#endif

    // Order-preserving keys for the 8 private elements.
    unsigned ua[8];
    ua[0] = order_key(__float_as_uint(a.x));
    ua[1] = order_key(__float_as_uint(a.y));
    ua[2] = order_key(__float_as_uint(a.z));
    ua[3] = order_key(__float_as_uint(a.w));
    ua[4] = order_key(__float_as_uint(b.x));
    ua[5] = order_key(__float_as_uint(b.y));
    ua[6] = order_key(__float_as_uint(b.z));
    ua[7] = order_key(__float_as_uint(b.w));

    // 4-pass MSB-first radix select of the K_RANK-th smallest key.
    unsigned k = K_RANK;
    unsigned prefix = 0u, pmask = 0u;
    for (int shift = 24; shift >= 0; shift -= 8) {
        hist[tid] = 0u;
        __syncthreads();
#pragma unroll
        for (int i = 0; i < 8; ++i) {
            if ((ua[i] & pmask) == prefix) {
                atomicAdd(&hist[(ua[i] >> shift) & 0xFFu], 1u);
            }
        }
        __syncthreads();

        // Two-level inclusive scan over the 256 bins:
        //   level 1: wave32 shuffle scan (no barriers)
        //   level 2: 8 wave totals via LDS + one barrier
        const unsigned h = hist[tid];
        unsigned s = h;
#pragma unroll
        for (int off = 1; off < 32; off <<= 1) {
            const unsigned t = __shfl_up(s, off, 32);
            if (lane >= off) s += t;
        }
        if (lane == 31) wtot[wave] = s;
        __syncthreads();
        unsigned base = 0u;
        for (int w = 0; w < wave; ++w) base += wtot[w];
        const unsigned incl = s + base;
        const unsigned excl = incl - h;
        if (k >= excl && k < incl) {  // exactly one thread matches
            sel[0] = (unsigned)tid;
            sel[1] = k - excl;
        }
        __syncthreads();
        prefix |= sel[0] << shift;
        k = sel[1];
        pmask |= 0xFFu << shift;
        // no trailing barrier needed: next sel/wtot writes are >=2 barriers away
    }

    // Invert the key map to get theta as a float; strict float compare
    // reproduces the reference mask exactly (incl. duplicate/-0 semantics).
    const unsigned tb = (prefix & 0x80000000u) ? (prefix ^ 0x80000000u) : ~prefix;
    const float theta = __uint_as_float(tb);

    float4 oa, ob;
    oa.x = (a.x > theta) ? a.x : 0.0f;
    oa.y = (a.y > theta) ? a.y : 0.0f;
    oa.z = (a.z > theta) ? a.z : 0.0f;
    oa.w = (a.w > theta) ? a.w : 0.0f;
    ob.x = (b.x > theta) ? b.x : 0.0f;
    ob.y = (b.y > theta) ? b.y : 0.0f;
    ob.z = (b.z > theta) ? b.z : 0.0f;
    ob.w = (b.w > theta) ? b.w : 0.0f;

    float4* orow = (float4*)(out + (size_t)row * DIM);
    orow[tid] = oa;
    orow[tid + BLOCK] = ob;
}

extern "C" void kernel_launch(void* const* d_in, const int* in_sizes, int n_in,
                              void* d_out, int out_size, void* d_ws, size_t ws_size,
                              hipStream_t stream) {
    const float* x = (const float*)d_in[0];
    float* out = (float*)d_out;
    const int rows = in_sizes[0] / DIM;   // 16384
    ksparse_rowselect<<<dim3(rows), dim3(BLOCK), 0, stream>>>(x, out);
}